// DecoderModel_77240691851867
// MI455X (gfx1250) — compile-verified
//
#include <hip/hip_runtime.h>
#include <math.h>

// Problem constants (fixed by the reference)
#define NNODES 10000      // N
#define NB     4          // batch
#define UNITS  128
#define CFEAT  129        // 1 + UNITS
#define KDIM   645        // CFEAT * 5 (GEMM reduction dim)
#define KPAD   648        // padded to multiple of 4 for wmma 16x16x4
#define KMAIN  644        // last guarded step starts here (161 unguarded steps)
#define FW     516        // CFEAT * NB  (row width of X_m matrices)
#define NROWS  40000      // NB * NNODES (GEMM rows)
#define NXF    (NNODES * FW)   // floats per X_m buffer

typedef __attribute__((ext_vector_type(2))) float v2f;
typedef __attribute__((ext_vector_type(8))) float v8f;

// ---------------------------------------------------------------------------
// small utility kernels
// ---------------------------------------------------------------------------
__global__ __launch_bounds__(256) void k_fill_zero(float* dst, int n) {
  int i = blockIdx.x * 256 + threadIdx.x;
  if (i < n) dst[i] = 0.0f;
}

__global__ __launch_bounds__(256) void k_scale_copy(float* dst, const float* __restrict__ src,
                                                    float scale, int n) {
  int i = blockIdx.x * 256 + threadIdx.x;
  if (i < n) dst[i] = scale * src[i];
}

__global__ __launch_bounds__(256) void k_degrees(const int* __restrict__ esrc,
                                                 const int* __restrict__ edst,
                                                 const float* __restrict__ ew,
                                                 float* deg_out, float* deg_in, int E) {
  int e = blockIdx.x * 256 + threadIdx.x;
  if (e < E) {
    atomicAdd(&deg_out[esrc[e]], ew[e]);
    atomicAdd(&deg_in[edst[e]], ew[e]);
  }
}

__global__ __launch_bounds__(256) void k_norm_w(const int* __restrict__ esrc,
                                                const int* __restrict__ edst,
                                                const float* __restrict__ ew,
                                                const float* __restrict__ deg_out,
                                                const float* __restrict__ deg_in,
                                                float* w_fwd, float* w_bwd, int E) {
  int e = blockIdx.x * 256 + threadIdx.x;
  if (e < E) {
    w_fwd[e] = ew[e] / deg_out[esrc[e]];
    w_bwd[e] = ew[e] / deg_in[edst[e]];
  }
}

// Build x0 for the gates gconv: X0[n*516 + c*4 + b] = cat(input, h)[b,n,c]
__global__ __launch_bounds__(256) void k_x0_gates(const float* __restrict__ inp,
                                                  const float* __restrict__ hid,
                                                  float* __restrict__ X0) {
  int i = blockIdx.x * 256 + threadIdx.x;
  if (i >= NXF) return;
  int f = i % FW, n = i / FW;
  int c = f >> 2, b = f & 3;
  X0[i] = (c == 0) ? inp[b * NNODES + n]
                   : hid[(b * NNODES + n) * UNITS + (c - 1)];
}

// Build x0 for the candidate gconv: channel c>=1 is r*h
__global__ __launch_bounds__(256) void k_x0_cand(const float* __restrict__ inp,
                                                 const float* __restrict__ hid,
                                                 const float* __restrict__ gates,
                                                 float* __restrict__ X0) {
  int i = blockIdx.x * 256 + threadIdx.x;
  if (i >= NXF) return;
  int f = i % FW, n = i / FW;
  int c = f >> 2, b = f & 3;
  if (c == 0) {
    X0[i] = inp[b * NNODES + n];
  } else {
    int row = b * NNODES + n;
    float r = gates[row * 256 + (c - 1)];          // reset gate (first 128 cols)
    float h = hid[row * UNITS + (c - 1)];
    X0[i] = r * h;
  }
}

// Y[rows[e], :] += coef * w[e] * X[cols[e], :]   (one wave per edge, 516 floats)
__global__ __launch_bounds__(256) void k_spmm(const int* __restrict__ rows,
                                              const int* __restrict__ cols,
                                              const float* __restrict__ w,
                                              const float* __restrict__ X,
                                              float* __restrict__ Y,
                                              float coef, int E) {
  int e = blockIdx.x * 8 + (threadIdx.x >> 5);
  if (e >= E) return;
  int lane = threadIdx.x & 31;
  int r = rows[e], c = cols[e];
  float cw = coef * w[e];
  const float* xin = X + (size_t)c * FW;
  float* yout = Y + (size_t)r * FW;
  for (int f = lane; f < FW; f += 32)
    atomicAdd(&yout[f], cw * xin[f]);
}

// ---------------------------------------------------------------------------
// WMMA GEMM core: one wave owns a 32-row x 128-col output block
// (two 16-row A tiles sharing every B fragment -> 16 WMMA per 20 loads).
// A is gathered on the fly from the 5 Chebyshev term buffers:
//   feature k = c*5 + m  ->  Xall[m*NXF + n*516 + c*4 + b],  row = b*N + n
// The (c,m) -> offset map is strength-reduced: k += 4 gives
//   m==0 ? (m=4,            off += 4*NXF)
//        : (m-=1, c+=1,     off += 4 - NXF)
// Lane layouts per CDNA5 ISA 7.12.2:
//   A 16x4 f32 : lane&15 = M, lane>=16 selects K+2, vgpr0/1 = K+0/K+1
//   B 4x16 f32 : lane&15 = N, lane>=16 selects K+2, vgpr0/1 = K+0/K+1
//   C 16x16 f32: lane&15 = N, lane>=16 -> M += 8, vgpr r = M row
// ---------------------------------------------------------------------------
#define WMMA_F32X4(A, B, C) __builtin_amdgcn_wmma_f32_16x16x4_f32( \
    false, (A), false, (B), (short)0, (C), false, false)

__device__ __forceinline__ void gemm_core2(const float* __restrict__ Xall,
                                           const float* __restrict__ W, int ldw,
                                           int row0, int colbase,
                                           v8f accA[8], v8f accB[8]) {
  const int lane = threadIdx.x & 31;
  const int lrow = lane & 15;
  const int koff = (lane >> 4) << 1;          // 0 or 2

  const int rowA = row0 + lrow;               // GEMM row = b*N + n
  const int rowB = row0 + 16 + lrow;
  const int bA = rowA / NNODES, nA = rowA - bA * NNODES;
  const int bB = rowB / NNODES, nB = rowB - bB * NNODES;
  const int rowoffA = nA * FW + bA;
  const int rowoffB = nB * FW + bB;

  // incremental (c,m) -> A offset state for this lane's two K slots
  int m0 = koff;                 // k0 = koff     (c = 0)
  int m1 = koff + 1;             // k1 = koff + 1 (c = 0)
  int aoff0 = m0 * NXF;
  int aoff1 = m1 * NXF;

  // ---- main loop: kb in [0, 644), no guards, no divisions ----
  for (int kb = 0; kb < KMAIN; kb += 4) {
    float a0A = Xall[rowoffA + aoff0];
    float a1A = Xall[rowoffA + aoff1];
    float a0B = Xall[rowoffB + aoff0];
    float a1B = Xall[rowoffB + aoff1];
    v2f Aa = {a0A, a1A};
    v2f Ab = {a0B, a1B};

    int kw0 = kb + koff;
    int kw1 = kw0 + 1;
#pragma unroll
    for (int t = 0; t < 8; ++t) {
      int col = colbase + t * 16 + lrow;
      v2f Bf = {W[(size_t)kw0 * ldw + col], W[(size_t)kw1 * ldw + col]};
      accA[t] = WMMA_F32X4(Aa, Bf, accA[t]);
      accB[t] = WMMA_F32X4(Ab, Bf, accB[t]);
    }

    // advance k by 4
    aoff0 += (m0 == 0) ? (4 * NXF) : (4 - NXF);
    m0 = (m0 == 0) ? 4 : (m0 - 1);
    aoff1 += (m1 == 0) ? (4 * NXF) : (4 - NXF);
    m1 = (m1 == 0) ? 4 : (m1 - 1);
  }

  // ---- guarded tail: kb = 644 (k = 644..647, only k=644 is real) ----
  {
    int k0 = KMAIN + koff;                    // 644 or 646
    int k1 = k0 + 1;                          // 645 or 647
    int kc0 = (k0 < KDIM) ? k0 : (KDIM - 1);  // clamp -> in-bounds address
    int kc1 = (k1 < KDIM) ? k1 : (KDIM - 1);
    int c0 = kc0 / 5, mm0 = kc0 - c0 * 5;
    int c1 = kc1 / 5, mm1 = kc1 - c1 * 5;

    float a0A = Xall[(size_t)mm0 * NXF + rowoffA + (c0 << 2)];
    float a1A = Xall[(size_t)mm1 * NXF + rowoffA + (c1 << 2)];
    float a0B = Xall[(size_t)mm0 * NXF + rowoffB + (c0 << 2)];
    float a1B = Xall[(size_t)mm1 * NXF + rowoffB + (c1 << 2)];
    if (k0 >= KDIM) { a0A = 0.0f; a0B = 0.0f; }
    if (k1 >= KDIM) { a1A = 0.0f; a1B = 0.0f; }
    v2f Aa = {a0A, a1A};
    v2f Ab = {a0B, a1B};

#pragma unroll
    for (int t = 0; t < 8; ++t) {
      int col = colbase + t * 16 + lrow;
      float b0 = W[(size_t)kc0 * ldw + col];
      float b1 = W[(size_t)kc1 * ldw + col];
      if (k0 >= KDIM) b0 = 0.0f;
      if (k1 >= KDIM) b1 = 0.0f;
      v2f Bf = {b0, b1};
      accA[t] = WMMA_F32X4(Aa, Bf, accA[t]);
      accB[t] = WMMA_F32X4(Ab, Bf, accB[t]);
    }
  }
}

// gates GEMM: (40000 x 645) x (645 x 256) -> sigmoid -> gates buffer [r | u]
__global__ __launch_bounds__(32) void k_gemm_gates(const float* __restrict__ Xall,
                                                   const float* __restrict__ Wg,
                                                   const float* __restrict__ bg,
                                                   float* __restrict__ gates) {
  int row0 = blockIdx.x * 32;
  int colbase = blockIdx.y * 128;
  v8f accA[8] = {};
  v8f accB[8] = {};
  gemm_core2(Xall, Wg, 2 * UNITS, row0, colbase, accA, accB);

  int lane = threadIdx.x & 31;
  int col_l = lane & 15;
  int rbase = (lane >> 4) * 8;
#pragma unroll
  for (int t = 0; t < 8; ++t) {
    int col = colbase + t * 16 + col_l;
    float bv = bg[col];
#pragma unroll
    for (int r = 0; r < 8; ++r) {
      int rowA = row0 + rbase + r;
      int rowB = rowA + 16;
      float vA = accA[t][r] + bv;
      float vB = accB[t][r] + bv;
      gates[(size_t)rowA * 256 + col] = 1.0f / (1.0f + __expf(-vA));
      gates[(size_t)rowB * 256 + col] = 1.0f / (1.0f + __expf(-vB));
    }
  }
}

// candidate GEMM: (40000 x 645) x (645 x 128) -> tanh -> fused GRU update
__global__ __launch_bounds__(32) void k_gemm_cand(const float* __restrict__ Xall,
                                                  const float* __restrict__ Wc,
                                                  const float* __restrict__ bc,
                                                  const float* __restrict__ gates,
                                                  const float* __restrict__ hid,
                                                  float* __restrict__ hidOut) {
  int row0 = blockIdx.x * 32;
  v8f accA[8] = {};
  v8f accB[8] = {};
  gemm_core2(Xall, Wc, UNITS, row0, 0, accA, accB);

  int lane = threadIdx.x & 31;
  int col_l = lane & 15;
  int rbase = (lane >> 4) * 8;
#pragma unroll
  for (int t = 0; t < 8; ++t) {
    int col = t * 16 + col_l;
    float bv = bc[col];
#pragma unroll
    for (int r = 0; r < 8; ++r) {
      int rowA = row0 + rbase + r;
      int rowB = rowA + 16;
      float cA = tanhf(accA[t][r] + bv);
      float cB = tanhf(accB[t][r] + bv);
      float uA = gates[(size_t)rowA * 256 + UNITS + col];
      float uB = gates[(size_t)rowB * 256 + UNITS + col];
      float hA = hid[(size_t)rowA * UNITS + col];
      float hB = hid[(size_t)rowB * UNITS + col];
      hidOut[(size_t)rowA * UNITS + col] = uA * hA + (1.0f - uA) * cA;
      hidOut[(size_t)rowB * UNITS + col] = uB * hB + (1.0f - uB) * cB;
    }
  }
}

// projection: out[row] = dot(new_h[row,:], W_proj) + b_proj  (wave per row)
__global__ __launch_bounds__(256) void k_proj(const float* __restrict__ hidOut,
                                              const float* __restrict__ Wp,
                                              const float* __restrict__ bp,
                                              float* __restrict__ out) {
  int row = blockIdx.x * 8 + (threadIdx.x >> 5);
  int lane = threadIdx.x & 31;
  float s = 0.0f;
#pragma unroll
  for (int i = 0; i < 4; ++i) {
    int u = lane * 4 + i;
    s += hidOut[(size_t)row * UNITS + u] * Wp[u];
  }
#pragma unroll
  for (int off = 16; off > 0; off >>= 1)
    s += __shfl_down(s, off, 32);
  if (lane == 0) out[row] = s + bp[0];
}

// ---------------------------------------------------------------------------
static void cheb_chain(const int* esrc, const int* edst,
                       const float* w_fwd, const float* w_bwd,
                       float* X0, float* X1f, float* X2f, float* X1b, float* X2b,
                       int E, hipStream_t stream) {
  dim3 blk(256);
  dim3 gE((E + 7) / 8);
  dim3 gI((NXF + 255) / 256);
  // forward support: x1 = S x0 ; x2 = 2 S x1 - x0
  k_fill_zero<<<gI, blk, 0, stream>>>(X1f, NXF);
  k_spmm<<<gE, blk, 0, stream>>>(esrc, edst, w_fwd, X0, X1f, 1.0f, E);
  k_scale_copy<<<gI, blk, 0, stream>>>(X2f, X0, -1.0f, NXF);
  k_spmm<<<gE, blk, 0, stream>>>(esrc, edst, w_fwd, X1f, X2f, 2.0f, E);
  // backward support
  k_fill_zero<<<gI, blk, 0, stream>>>(X1b, NXF);
  k_spmm<<<gE, blk, 0, stream>>>(edst, esrc, w_bwd, X0, X1b, 1.0f, E);
  k_scale_copy<<<gI, blk, 0, stream>>>(X2b, X0, -1.0f, NXF);
  k_spmm<<<gE, blk, 0, stream>>>(edst, esrc, w_bwd, X1b, X2b, 2.0f, E);
}

extern "C" void kernel_launch(void* const* d_in, const int* in_sizes, int n_in,
                              void* d_out, int out_size, void* d_ws, size_t ws_size,
                              hipStream_t stream) {
  const float* inp  = (const float*)d_in[0];   // (4, 10000)
  const float* hid  = (const float*)d_in[1];   // (1, 4, 10000*128)
  const int*   esrc = (const int*)d_in[2];     // (E,)
  const int*   edst = (const int*)d_in[3];
  const float* ew   = (const float*)d_in[4];
  const float* Wg   = (const float*)d_in[5];   // (645, 256)
  const float* bg   = (const float*)d_in[6];   // (256,)
  const float* Wc   = (const float*)d_in[7];   // (645, 128)
  const float* bc   = (const float*)d_in[8];   // (128,)
  const float* Wp   = (const float*)d_in[9];   // (128,)
  const float* bp   = (const float*)d_in[10];  // (1,)
  const int E = in_sizes[2];

  float* out    = (float*)d_out;               // [output 40000 | hidden 5120000]
  float* hidOut = out + NROWS;

  // workspace carving (~139 MiB total)
  float* ws      = (float*)d_ws;
  float* deg_out = ws;                          // N
  float* deg_in  = deg_out + NNODES;            // N
  float* w_fwd   = deg_in + NNODES;             // E
  float* w_bwd   = w_fwd + E;                   // E
  float* Xall    = w_bwd + E;                   // 5 * NXF (x0,x1f,x2f,x1b,x2b)
  float* gates   = Xall + 5 * (size_t)NXF;      // NROWS * 256
  (void)n_in; (void)out_size; (void)ws_size;

  float* X0  = Xall;
  float* X1f = Xall + 1 * (size_t)NXF;
  float* X2f = Xall + 2 * (size_t)NXF;
  float* X1b = Xall + 3 * (size_t)NXF;
  float* X2b = Xall + 4 * (size_t)NXF;

  dim3 blk(256);

  // 1) dual random-walk support weights
  k_fill_zero<<<dim3((2 * NNODES + 255) / 256), blk, 0, stream>>>(deg_out, 2 * NNODES);
  k_degrees<<<dim3((E + 255) / 256), blk, 0, stream>>>(esrc, edst, ew, deg_out, deg_in, E);
  k_norm_w<<<dim3((E + 255) / 256), blk, 0, stream>>>(esrc, edst, ew, deg_out, deg_in,
                                                      w_fwd, w_bwd, E);

  // 2) gconv #1 -> gates
  k_x0_gates<<<dim3((NXF + 255) / 256), blk, 0, stream>>>(inp, hid, X0);
  cheb_chain(esrc, edst, w_fwd, w_bwd, X0, X1f, X2f, X1b, X2b, E, stream);
  k_gemm_gates<<<dim3(NROWS / 32, 2), dim3(32), 0, stream>>>(Xall, Wg, bg, gates);

  // 3) gconv #2 -> candidate (+ fused GRU update into hidden output)
  k_x0_cand<<<dim3((NXF + 255) / 256), blk, 0, stream>>>(inp, hid, gates, X0);
  cheb_chain(esrc, edst, w_fwd, w_bwd, X0, X1f, X2f, X1b, X2b, E, stream);
  k_gemm_cand<<<dim3(NROWS / 32, 1), dim3(32), 0, stream>>>(Xall, Wc, bc, gates, hid, hidOut);

  // 4) output projection
  k_proj<<<dim3(NROWS / 8), blk, 0, stream>>>(hidOut, Wp, bp, out);
}